// HiddenState_DB_34720515621652
// MI455X (gfx1250) — compile-verified
//
#include <hip/hip_runtime.h>
#include <hip/hip_bf16.h>

typedef __attribute__((ext_vector_type(2))) float v2f;
typedef __attribute__((ext_vector_type(8))) float v8f;

#if __has_builtin(__builtin_amdgcn_tanhf)
#define TANH_F32(v) __builtin_amdgcn_tanhf(v)
#else
#define TANH_F32(v) tanhf(v)
#endif

// Problem constants (from setup_inputs): B=8, T=32, Fin=64, H=64, K=5, E=1, N=512
#define B_   8
#define T_   32
#define FIN_ 64
#define H_   64
#define N_   512
#define K2_HSTRIDE 320   // E*K*Fin = 1*5*64
#define LROW 20          // padded LDS row stride (floats) -> conflict-free halves

// One wave handles 16 columns (n-tile) of one batch: full 64-row GEMM chain.
// A-frag layout (16x4 f32): lane m = lane%16, k = 4c + 2*(lane/16) + v
// B-frag layout (4x16 f32): lane n = lane%16, k = 4c + 2*(lane/16) + v
// D layout   (16x16 f32):   reg r, lane: m = 16*mt + r + 8*(lane/16), n = lane%16
__global__ __launch_bounds__(128) void grnn_db_kernel(
    const float* __restrict__ x,      // (B,T,Fin,N)
    const float* __restrict__ K1,     // (H,H)
    const float* __restrict__ K2,     // (H,E,K,Fin)
    const float* __restrict__ xBias,  // (H,1)
    const float* __restrict__ wBias,  // (H,1)
    float* __restrict__ out)          // z (B,T,H,N) ++ zT (B,H,N)
{
  __shared__ float lds[4][H_ * LROW];

  const int lane = threadIdx.x & 31;
  const int wid  = threadIdx.x >> 5;
  const int gw   = blockIdx.x * 4 + wid;   // 0..255
  const int b    = gw >> 5;                // batch
  const int n0   = (gw & 31) * 16;         // column base
  const int nl   = lane & 15;              // n within tile
  const int hl   = lane >> 4;              // lane half

  float* wbuf = lds[wid];                  // wave-private 64x16 (padded)

  // ---- K1 A-fragments resident in registers for the whole recurrence ----
  v2f a1[4][16];
#pragma unroll
  for (int mt = 0; mt < 4; ++mt) {
    const int m = 16 * mt + nl;
#pragma unroll
    for (int c = 0; c < 16; ++c) {
      const int k = 4 * c + 2 * hl;
      a1[mt][c] = *(const v2f*)(K1 + m * H_ + k);   // (K1[m][k], K1[m][k+1])
    }
  }

  // ---- w0 = K2[:,0,0,:] @ x[b,0,:,n-tile]  (16 chained K=4 WMMAs / tile) ----
  v8f acc[4];
#pragma unroll
  for (int mt = 0; mt < 4; ++mt) acc[mt] = (v8f)0.0f;

  const float* xb = x + (size_t)b * T_ * FIN_ * N_ + n0 + nl;  // t=0 slice
#pragma unroll
  for (int c = 0; c < 16; ++c) {
    const int k = 4 * c + 2 * hl;
    v2f bf;
    bf.x = xb[(k + 0) * N_];
    bf.y = xb[(k + 1) * N_];
#pragma unroll
    for (int mt = 0; mt < 4; ++mt) {
      const int m = 16 * mt + nl;
      v2f af = *(const v2f*)(K2 + m * K2_HSTRIDE + k);  // K2[m,0,0,k..k+1]
      acc[mt] = __builtin_amdgcn_wmma_f32_16x16x4_f32(
          false, af, false, bf, (short)0, acc[mt], false, false);
    }
  }

  // ---- add xBias; emit t=0; seed LDS; build comb = w0 + wBias ----
  float comb[4][8];
  float* out_z = out + ((size_t)b * T_) * H_ * N_ + n0 + nl;
#pragma unroll
  for (int mt = 0; mt < 4; ++mt) {
#pragma unroll
    for (int r = 0; r < 8; ++r) {
      const int m = 16 * mt + r + 8 * hl;
      const float v = acc[mt][r] + xBias[m];
      out_z[(size_t)m * N_] = v;          // z[b,0,m,n]
      wbuf[m * LROW + nl] = v;            // running w for next step
      comb[mt][r] = v + wBias[m];         // fixed additive term per step
    }
  }

  // ---- recurrence: w_t = tanh(K1 @ w_{t-1} + comb) ----
  for (int t = 1; t < T_; ++t) {
    v8f d[4];
#pragma unroll
    for (int mt = 0; mt < 4; ++mt)
#pragma unroll
      for (int r = 0; r < 8; ++r) d[mt][r] = comb[mt][r];

#pragma unroll
    for (int c = 0; c < 16; ++c) {
      const int k = 4 * c + 2 * hl;
      v2f bf;
      bf.x = wbuf[(k + 0) * LROW + nl];
      bf.y = wbuf[(k + 1) * LROW + nl];
#pragma unroll
      for (int mt = 0; mt < 4; ++mt)
        d[mt] = __builtin_amdgcn_wmma_f32_16x16x4_f32(
            false, a1[mt][c], false, bf, (short)0, d[mt], false, false);
    }

    float* out_t = out + ((size_t)b * T_ + t) * H_ * N_ + n0 + nl;
#pragma unroll
    for (int mt = 0; mt < 4; ++mt) {
#pragma unroll
      for (int r = 0; r < 8; ++r) {
        const int m = 16 * mt + r + 8 * hl;
        const float v = TANH_F32(d[mt][r]);
        out_t[(size_t)m * N_] = v;        // z[b,t,m,n]
        wbuf[m * LROW + nl] = v;          // same-wave DS ops are in-order: safe
        if (t == T_ - 1) {                // zT = z[:, -1]
          out[(size_t)B_ * T_ * H_ * N_ + ((size_t)b * H_ + m) * N_ + n0 + nl] = v;
        }
      }
    }
  }
}

extern "C" void kernel_launch(void* const* d_in, const int* in_sizes, int n_in,
                              void* d_out, int out_size, void* d_ws, size_t ws_size,
                              hipStream_t stream) {
  // setup_inputs order: x, z0, S, K1, K2, xBias, wBias  (z0 and S are unused by
  // the reference computation -- only x[:,0] feeds the recurrence).
  const float* x     = (const float*)d_in[0];
  const float* K1    = (const float*)d_in[3];
  const float* K2    = (const float*)d_in[4];
  const float* xBias = (const float*)d_in[5];
  const float* wBias = (const float*)d_in[6];
  float* out = (float*)d_out;

  // 256 waves total = B(8) * N/16(32); 4 waves per block.
  dim3 grid(64), block(128);
  hipLaunchKernelGGL(grnn_db_kernel, grid, block, 0, stream,
                     x, K1, K2, xBias, wBias, out);
}